// SplineGCN_20839181320473
// MI455X (gfx1250) — compile-verified
//
#include <hip/hip_runtime.h>

#define N_NODES   100000
#define N_EDGES   1000000
#define F_IN      32
#define F_OUT     32
#define K_KERNEL  25
#define RDIM      (K_KERNEL * F_IN)   // 800: flattened (k,i) reduction dim

typedef float v2f __attribute__((ext_vector_type(2)));
typedef float v8f __attribute__((ext_vector_type(8)));

#if defined(__has_builtin)
#if __has_builtin(__builtin_amdgcn_global_load_async_to_lds_b128)
#define HAVE_ASYNC_LDS 1
#endif
#endif

#ifdef HAVE_ASYNC_LDS
// builtin signature (from hipcc diagnostic): param0 = AS1 int-vec16*, etc.
typedef int v4i_vs __attribute__((vector_size(16)));
typedef __attribute__((address_space(1))) v4i_vs* async_gptr_t;
typedef __attribute__((address_space(3))) v4i_vs* async_lptr_t;
#define WAIT_ASYNCCNT_5() asm volatile("s_wait_asynccnt 0x5" ::: "memory")
#define WAIT_ASYNCCNT_0() asm volatile("s_wait_asynccnt 0x0" ::: "memory")
#else
#define WAIT_ASYNCCNT_5()
#define WAIT_ASYNCCNT_0()
#endif

// ---------------------------------------------------------------------------
// Phase 1: per-edge spline basis + feature gather + scatter into agg buckets.
// One wave32 per edge batch; lane == input-feature channel (F_IN == 32).
// ---------------------------------------------------------------------------
__global__ void __launch_bounds__(256)
spline_scatter(const float* __restrict__ features,
               const float* __restrict__ pseudo,
               const int*   __restrict__ edge_index,
               float*       __restrict__ agg,
               float*       __restrict__ deg) {
  const int lane   = threadIdx.x & 31;
  const int waveId = blockIdx.x * (blockDim.x >> 5) + (threadIdx.x >> 5);
  const int nWaves = gridDim.x * (blockDim.x >> 5);

  for (int e = waveId; e < N_EDGES; e += nWaves) {
    const int   row = edge_index[e];             // target
    const int   col = edge_index[N_EDGES + e];   // source
    const float p0  = pseudo[2 * e + 0];
    const float p1  = pseudo[2 * e + 1];

    // degree-1 open B-spline basis on a 5x5 grid: u in [0,4]
    const float u0 = p0 * 4.0f, u1 = p1 * 4.0f;
    float f0 = fminf(fmaxf(floorf(u0), 0.0f), 4.0f);
    float f1 = fminf(fmaxf(floorf(u1), 0.0f), 4.0f);
    const float fr0 = u0 - f0, fr1 = u1 - f1;
    const int i00 = (int)f0, i01 = (int)f1;
    const int i10 = min(i00 + 1, 4), i11 = min(i01 + 1, 4);

    const float x = features[col * F_IN + lane]; // coalesced 128B gather

#pragma unroll
    for (int s = 0; s < 4; ++s) {
      const int   b0 = s & 1;          // bit for dim 0
      const int   b1 = (s >> 1) & 1;   // bit for dim 1
      const int   k0 = b0 ? i10 : i00;
      const int   k1 = b1 ? i11 : i01;
      const float w  = (b0 ? fr0 : (1.0f - fr0)) * (b1 ? fr1 : (1.0f - fr1));
      const int   kidx = k0 * 5 + k1;  // strides [5,1]
      unsafeAtomicAdd(&agg[((size_t)row * K_KERNEL + kidx) * F_IN + lane], w * x);
    }
    if (lane == 0) unsafeAtomicAdd(&deg[row], 1.0f);
  }
}

// ---------------------------------------------------------------------------
// Phase 2: out[n,o] = (sum_r agg[n,r] * W[r,o]) / max(deg[n],1) + bias[o]
// GEMM [N x 800] x [800 x 32] with V_WMMA_F32_16X16X4_F32.
// 8 waves/WG, 16 nodes/wave -> 128 nodes/WG.  Double-buffered LDS pipeline:
// A tiles streamed with GLOBAL_LOAD_ASYNC_TO_LDS_B128 (ASYNCcnt), W chunks
// pre-swizzled so every B operand is one conflict-free ds_load_b64.
// ---------------------------------------------------------------------------
#define WG_THREADS      256
#define WAVES_PER_WG    8
#define NODES_PER_WAVE  16
#define NODES_PER_WG    (WAVES_PER_WG * NODES_PER_WAVE)   // 128
#define CHUNK_K         40                                // 800 = 20 chunks
#define NUM_CHUNKS      (RDIM / CHUNK_K)                  // 20
#define KGROUPS         (CHUNK_K / 4)                     // 10 wmma k-groups
#define LDS_A_STRIDE    44    // pad: 44*4B=176B (16B-aligned rows, 0-conflict)
#define A_FLOATS        (NODES_PER_WG * LDS_A_STRIDE)     // 5632
#define W_FLOATS        (KGROUPS * 128)                   // 1280 (swizzled)

__global__ void __launch_bounds__(WG_THREADS)
spline_gemm(const float* __restrict__ agg,
            const float* __restrict__ weight,   // [800,32] flat (k*32+i, o)
            const float* __restrict__ bias,
            const float* __restrict__ deg,
            float*       __restrict__ out) {
  __shared__ float ldsA[2][A_FLOATS];   // 2 x 22528 B
  __shared__ float ldsW[2][W_FLOATS];   // 2 x  5120 B

  const int tid   = threadIdx.x;
  const int lane  = tid & 31;
  const int wave  = tid >> 5;
  const int l     = lane & 15;   // M-row-within-half / N column
  const int hi    = lane >> 4;   // lane-half selector
  const int node0 = blockIdx.x * NODES_PER_WG;

  v8f acc0 = {};  // outputs 0..15
  v8f acc1 = {};  // outputs 16..31

  // ---- staging helpers (exact trip counts: no divergence) ----
  auto stageA = [&](int c, int buf) {
    const int rbase = c * CHUNK_K;
#pragma unroll
    for (int k = 0; k < 5; ++k) {
      const int idx16 = tid + k * WG_THREADS;       // 0..1279 (16B units)
      const int i     = idx16 / 10;                 // node row 0..127
      const int j16   = idx16 - i * 10;             // 0..9 (x16B = 40 floats)
      int n = node0 + i;
      if (n >= N_NODES) n = N_NODES - 1;            // clamp (tail-safe reads)
      const float* g = agg + (size_t)n * RDIM + rbase + j16 * 4;
      float*       p = &ldsA[buf][i * LDS_A_STRIDE + j16 * 4];
#ifdef HAVE_ASYNC_LDS
      __builtin_amdgcn_global_load_async_to_lds_b128(
          (async_gptr_t)g, (async_lptr_t)p, 0, 0);
#else
      *(float4*)p = *(const float4*)g;
#endif
    }
  };

  auto stageW = [&](int c, int buf) {
    const int rbase = c * CHUNK_K;
#pragma unroll
    for (int k = 0; k < 5; ++k) {
      const int idx = tid + k * WG_THREADS;         // 0..1279
      const int rl  = idx >> 5;                     // local r 0..39
      const int o   = idx & 31;                     // output col
      const float v = weight[(rbase + rl) * F_OUT + o];
      // swizzle: lane L of group g reads {W[4g+2hi][o], W[4g+2hi+1][o]}
      // at floats [g*128 + (o>>4)*64 + L*2 + comp]
      const int g    = rl >> 2;
      const int rr   = rl & 3;
      const int dest = g * 128 + (o >> 4) * 64 +
                       (((rr >> 1) * 16 + (o & 15)) << 1) + (rr & 1);
      ldsW[buf][dest] = v;
    }
  };

  // ---- software pipeline: prefetch chunk c+1 while computing chunk c ----
  stageA(0, 0);
  stageW(0, 0);

  for (int c = 0; c < NUM_CHUNKS; ++c) {
    const int cur = c & 1;
    if (c + 1 < NUM_CHUNKS) {
      stageA(c + 1, cur ^ 1);
      stageW(c + 1, cur ^ 1);
      WAIT_ASYNCCNT_5();    // our 5 copies for chunk c have landed
    } else {
      WAIT_ASYNCCNT_0();
    }
    __syncthreads();

    const float* Ab = &ldsA[cur][(wave * NODES_PER_WAVE + l) * LDS_A_STRIDE];
    const float* Wb = &ldsW[cur][0];
#pragma unroll
    for (int g = 0; g < KGROUPS; ++g) {
      const int kk = g * 4;
      // A 16x4: VGPR0 = K{0|2}, VGPR1 = K{1|3} per lane half
      const v2f a  = *(const v2f*)&Ab[kk + 2 * hi];
      // B 4x16: swizzled so each lane's pair is contiguous -> ds_load_b64
      const float* Wk = &Wb[g * 128];
      const v2f b0 = *(const v2f*)&Wk[lane * 2];        // outs 0..15
      const v2f b1 = *(const v2f*)&Wk[64 + lane * 2];   // outs 16..31

      acc0 = __builtin_amdgcn_wmma_f32_16x16x4_f32(
          false, a, false, b0, (short)0, acc0, false, false);
      acc1 = __builtin_amdgcn_wmma_f32_16x16x4_f32(
          false, a, false, b1, (short)0, acc1, false, false);
    }
    __syncthreads();
  }

  // Epilogue: degree-normalize + bias.  C/D layout: M = v + 8*hi, N = l.
  const float bia0 = bias[l];
  const float bia1 = bias[16 + l];
#pragma unroll
  for (int v = 0; v < 8; ++v) {
    const int node = node0 + wave * NODES_PER_WAVE + v + 8 * hi;
    if (node < N_NODES) {
      const float inv = 1.0f / fmaxf(deg[node], 1.0f);
      out[node * F_OUT + l]      = acc0[v] * inv + bia0;
      out[node * F_OUT + 16 + l] = acc1[v] * inv + bia1;
    }
  }
}

// ---------------------------------------------------------------------------
extern "C" void kernel_launch(void* const* d_in, const int* in_sizes, int n_in,
                              void* d_out, int out_size, void* d_ws, size_t ws_size,
                              hipStream_t stream) {
  const float* features  = (const float*)d_in[0];  // [100000, 32]
  const float* weight    = (const float*)d_in[1];  // [25, 32, 32]
  const float* bias      = (const float*)d_in[2];  // [32]
  const float* pseudo    = (const float*)d_in[3];  // [1000000, 2]
  const int*   edge_idx  = (const int*)d_in[4];    // [2, 1000000]
  float*       out       = (float*)d_out;          // [100000, 32]

  float* agg = (float*)d_ws;                         // 100000*800 floats
  float* deg = agg + (size_t)N_NODES * RDIM;         // 100000 floats

  const size_t zero_bytes = ((size_t)N_NODES * RDIM + N_NODES) * sizeof(float);
  (void)hipMemsetAsync(d_ws, 0, zero_bytes, stream); // graph-capturable

  spline_scatter<<<2048, 256, 0, stream>>>(features, pseudo, edge_idx, agg, deg);

  const int gemm_blocks = (N_NODES + NODES_PER_WG - 1) / NODES_PER_WG;
  spline_gemm<<<gemm_blocks, WG_THREADS, 0, stream>>>(agg, weight, bias, deg, out);
}